// GraphSAGEEmbModel_74491912782413
// MI455X (gfx1250) — compile-verified
//
#include <hip/hip_runtime.h>
#include <hip/hip_bf16.h>

typedef __attribute__((ext_vector_type(2))) float v2f;
typedef __attribute__((ext_vector_type(8))) float v8f;

#define HID 64

// ---------------- degree ----------------
__global__ void degree_kernel(const int* __restrict__ dst, float* __restrict__ deg, int E) {
    int e = blockIdx.x * blockDim.x + threadIdx.x;
    if (e < E) atomicAdd(&deg[dst[e]], 1.0f);
}

__global__ void deginv_kernel(float* __restrict__ deg, int N) {
    int i = blockIdx.x * blockDim.x + threadIdx.x;
    if (i < N) {
        float d = deg[i];
        deg[i] = (d > 0.0f) ? (1.0f / d) : 0.0f;   // deg>=1 when >0, matches 1/max(deg,1)
    }
}

// ---------------- embedding gather (float4 vectorized) ----------------
__global__ void gather_kernel(const int* __restrict__ node_ids,
                              const float4* __restrict__ emb4,
                              float4* __restrict__ h0v, int N) {
    long long t = (long long)blockIdx.x * blockDim.x + threadIdx.x;
    long long total = (long long)N * 8;            // 32 floats = 8 float4 per row
    if (t >= total) return;
    int i = (int)(t >> 3), q = (int)(t & 7);
    h0v[t] = emb4[(long long)node_ids[i] * 8 + q];
}

// ---------------- edge scatter-add: agg[dst] += h[src] ----------------
template<int W4>   // row width in float4 units (8 -> 32 floats, 16 -> 64 floats)
__global__ void scatter_kernel(const int* __restrict__ src, const int* __restrict__ dst,
                               const float4* __restrict__ hv, float* __restrict__ agg, int E) {
    long long t = (long long)blockIdx.x * blockDim.x + threadIdx.x;
    long long total = (long long)E * W4;
    if (t >= total) return;
    int e = (int)(t / W4), q = (int)(t % W4);
    int s = src[e], d = dst[e];
    float4 v = hv[(long long)s * W4 + q];
    float* base = &agg[(long long)d * (W4 * 4) + q * 4];
    atomicAdd(base + 0, v.x);
    atomicAdd(base + 1, v.y);
    atomicAdd(base + 2, v.z);
    atomicAdd(base + 3, v.w);
}

// ---------------- fused SAGE layer: relu(Hin@Wself + (Agg*dinv)@Wneigh + b) ----------------
// One block = 128 threads = 4 waves; block handles 32 rows (two 16-row tiles);
// wave w handles cols [16w,16w+16). Uses V_WMMA_F32_16X16X4_F32 (exact fp32 on the
// matrix pipe). Out-of-range rows are CLAMPED, not predicated: a clamped A row only
// feeds an out-of-range D row which is never stored, so EXEC stays uniform and the
// inner loop is branch-free (WMMA requires EXEC all-ones anyway).
template<int KDIM>
__global__ void __launch_bounds__(128)
sage_gemm_kernel(const float* __restrict__ Hin, const float* __restrict__ Agg,
                 const float* __restrict__ dinv,
                 const float* __restrict__ Wself, const float* __restrict__ Wneigh,
                 const float* __restrict__ bias, float* __restrict__ Hout, int N) {
    const int tid  = threadIdx.x;
    const int wave = tid >> 5;
    const int lane = tid & 31;
    const int l16  = lane & 15;
    const int hi   = lane >> 4;        // 0 or 1: selects K/M sub-half per ISA layout
    const int rowbase = blockIdx.x * 32;
    const int colbase = wave * 16;

    // clamped A-row indices for the two row tiles
    const int arow0 = min(rowbase + l16,      N - 1);
    const int arow1 = min(rowbase + 16 + l16, N - 1);
    const float dv0 = dinv[arow0];
    const float dv1 = dinv[arow1];

    const float* __restrict__ hp0 = Hin + (long long)arow0 * KDIM;
    const float* __restrict__ hp1 = Hin + (long long)arow1 * KDIM;
    const float* __restrict__ gp0 = Agg + (long long)arow0 * KDIM;
    const float* __restrict__ gp1 = Agg + (long long)arow1 * KDIM;

    v8f c0 = {0.f, 0.f, 0.f, 0.f, 0.f, 0.f, 0.f, 0.f};
    v8f c1 = {0.f, 0.f, 0.f, 0.f, 0.f, 0.f, 0.f, 0.f};

#pragma unroll
    for (int k = 0; k < KDIM; k += 4) {
        const int ak = k + hi * 2;     // lane holds A[row][ak], A[row][ak+1]
        // B fragments: B[ak][bcol], B[ak+1][bcol] — loaded once, used by both row tiles
        const int bcol = colbase + l16;
        v2f bS, bG;
        bS.x = Wself [(long long)ak       * HID + bcol];
        bS.y = Wself [(long long)(ak + 1) * HID + bcol];
        bG.x = Wneigh[(long long)ak       * HID + bcol];
        bG.y = Wneigh[(long long)(ak + 1) * HID + bcol];

        // A fragments (vector float2 loads; mean-normalization folded into agg tile)
        v2f aS0 = *(const v2f*)(hp0 + ak);
        v2f aS1 = *(const v2f*)(hp1 + ak);
        v2f aG0 = *(const v2f*)(gp0 + ak);
        v2f aG1 = *(const v2f*)(gp1 + ak);
        aG0.x *= dv0; aG0.y *= dv0;
        aG1.x *= dv1; aG1.y *= dv1;

        c0 = __builtin_amdgcn_wmma_f32_16x16x4_f32(false, aS0, false, bS, (short)0, c0, false, false);
        c1 = __builtin_amdgcn_wmma_f32_16x16x4_f32(false, aS1, false, bS, (short)0, c1, false, false);
        c0 = __builtin_amdgcn_wmma_f32_16x16x4_f32(false, aG0, false, bG, (short)0, c0, false, false);
        c1 = __builtin_amdgcn_wmma_f32_16x16x4_f32(false, aG1, false, bG, (short)0, c1, false, false);
    }

    const int col = colbase + l16;
    const float bv = bias[col];
#pragma unroll
    for (int r = 0; r < 8; ++r) {
        // C/D layout: VGPR r -> M=r (lanes 0-15), M=r+8 (lanes 16-31); N = lane&15
        const int m = r + (hi << 3);
        int row0 = rowbase + m;
        int row1 = rowbase + 16 + m;
        if (row0 < N) {
            float v = c0[r] + bv;
            Hout[(long long)row0 * HID + col] = v > 0.0f ? v : 0.0f;
        }
        if (row1 < N) {
            float v = c1[r] + bv;
            Hout[(long long)row1 * HID + col] = v > 0.0f ? v : 0.0f;
        }
    }
}

// ---------------- per-graph sum pooling ----------------
__global__ void pool_kernel(const int* __restrict__ gid, const float4* __restrict__ h2v,
                            float* __restrict__ hgsum, float* __restrict__ cnt, int N) {
    long long t = (long long)blockIdx.x * blockDim.x + threadIdx.x;
    long long total = (long long)N * 16;           // 64 floats = 16 float4 per row
    if (t >= total) return;
    int i = (int)(t >> 4), q = (int)(t & 15);
    int g = gid[i];
    float4 v = h2v[t];
    float* base = &hgsum[(long long)g * HID + q * 4];
    atomicAdd(base + 0, v.x);
    atomicAdd(base + 1, v.y);
    atomicAdd(base + 2, v.z);
    atomicAdd(base + 3, v.w);
    if (q == 0) atomicAdd(&cnt[g], 1.0f);
}

// ---------------- scorer MLP: relu(hg@Ws1+bs1)@Ws2+bs2 ----------------
__global__ void __launch_bounds__(64)
scorer_kernel(const float* __restrict__ hgsum, const float* __restrict__ cnt,
              const float* __restrict__ Ws1, const float* __restrict__ bs1,
              const float* __restrict__ Ws2, const float* __restrict__ bs2,
              float* __restrict__ out) {
    __shared__ float row[HID];
    __shared__ float t1[HID];
    const int b = blockIdx.x;
    const int j = threadIdx.x;
    const float inv = 1.0f / fmaxf(cnt[b], 1.0f);
    row[j] = hgsum[(long long)b * HID + j] * inv;
    __syncthreads();
    float acc = bs1[j];
#pragma unroll
    for (int k = 0; k < HID; ++k) acc += row[k] * Ws1[k * HID + j];
    t1[j] = fmaxf(acc, 0.0f) * Ws2[j];
    __syncthreads();
    for (int s = 32; s > 0; s >>= 1) {
        if (j < s) t1[j] += t1[j + s];
        __syncthreads();
    }
    if (j == 0) out[b] = t1[0] + bs2[0];
}

static inline size_t align_up(size_t x, size_t a) { return (x + a - 1) / a * a; }

extern "C" void kernel_launch(void* const* d_in, const int* in_sizes, int n_in,
                              void* d_out, int out_size, void* d_ws, size_t ws_size,
                              hipStream_t stream) {
    (void)n_in; (void)ws_size;
    const int*   node_ids = (const int*)  d_in[0];
    const int*   src      = (const int*)  d_in[1];
    const int*   dst      = (const int*)  d_in[2];
    const int*   gid      = (const int*)  d_in[3];
    const float* emb      = (const float*)d_in[4];
    const float* Wself1   = (const float*)d_in[5];
    const float* Wneigh1  = (const float*)d_in[6];
    const float* b1       = (const float*)d_in[7];
    const float* Wself2   = (const float*)d_in[8];
    const float* Wneigh2  = (const float*)d_in[9];
    const float* b2       = (const float*)d_in[10];
    const float* Ws1      = (const float*)d_in[11];
    const float* bs1      = (const float*)d_in[12];
    const float* Ws2      = (const float*)d_in[13];
    const float* bs2      = (const float*)d_in[14];

    const int N = in_sizes[0];
    const int E = in_sizes[1];
    const int B = out_size;
    float* out = (float*)d_out;

    // workspace layout (floats); accumulators first so one memset zeroes them all
    float* ws = (float*)d_ws;
    size_t off = 0;
    float* deg   = ws + off; off = align_up(off + (size_t)N,        64);
    float* agg1  = ws + off; off = align_up(off + (size_t)N * 32,   64);
    float* agg2  = ws + off; off = align_up(off + (size_t)N * 64,   64);
    float* hgsum = ws + off; off = align_up(off + (size_t)B * HID,  64);
    float* cnt   = ws + off; off = align_up(off + (size_t)B,        64);
    const size_t zero_floats = off;
    float* h0    = ws + off; off = align_up(off + (size_t)N * 32,   64);
    float* h1    = ws + off; off = align_up(off + (size_t)N * 64,   64);
    float* h2    = ws + off; off = align_up(off + (size_t)N * 64,   64);

    hipMemsetAsync(d_ws, 0, zero_floats * sizeof(float), stream);

    const int BT = 256;
    // degrees + reciprocal
    degree_kernel<<<(E + BT - 1) / BT, BT, 0, stream>>>(dst, deg, E);
    deginv_kernel<<<(N + BT - 1) / BT, BT, 0, stream>>>(deg, N);

    // embedding gather -> h0 [N,32]
    {
        long long total = (long long)N * 8;
        gather_kernel<<<(unsigned)((total + BT - 1) / BT), BT, 0, stream>>>(
            node_ids, (const float4*)emb, (float4*)h0, N);
    }

    // layer 1: scatter messages, then fused WMMA GEMM + bias + relu -> h1 [N,64]
    {
        long long total = (long long)E * 8;
        scatter_kernel<8><<<(unsigned)((total + BT - 1) / BT), BT, 0, stream>>>(
            src, dst, (const float4*)h0, agg1, E);
    }
    sage_gemm_kernel<32><<<(N + 31) / 32, 128, 0, stream>>>(
        h0, agg1, deg, Wself1, Wneigh1, b1, h1, N);

    // layer 2: scatter messages on h1, fused WMMA GEMM -> h2 [N,64]
    {
        long long total = (long long)E * 16;
        scatter_kernel<16><<<(unsigned)((total + BT - 1) / BT), BT, 0, stream>>>(
            src, dst, (const float4*)h1, agg2, E);
    }
    sage_gemm_kernel<64><<<(N + 31) / 32, 128, 0, stream>>>(
        h1, agg2, deg, Wself2, Wneigh2, b2, h2, N);

    // per-graph mean pooling + scorer MLP
    {
        long long total = (long long)N * 16;
        pool_kernel<<<(unsigned)((total + BT - 1) / BT), BT, 0, stream>>>(
            gid, (const float4*)h2, hgsum, cnt, N);
    }
    scorer_kernel<<<B, 64, 0, stream>>>(hgsum, cnt, Ws1, bs1, Ws2, bs2, out);
}